// Embedding_Table_Cache_Group_67396626809222
// MI455X (gfx1250) — compile-verified
//
#include <hip/hip_runtime.h>
#include <stdint.h>

// ---- problem constants (from the reference) ----
#define CACHE_SIZE 250007
#define NUM_WAYS   4
#define BATCH      4096
#define POOL       20
#define DIM        64
#define NTAB       2

// Async store-from-LDS path (gfx1250 ASYNCcnt engine), with graceful fallback.
#if defined(__has_builtin)
#  if __has_builtin(__builtin_amdgcn_global_store_async_from_lds_b64)
#    define HAVE_ASYNC_STORE 1
#  endif
#endif
#ifndef HAVE_ASYNC_STORE
#  define HAVE_ASYNC_STORE 0
#endif

#if HAVE_ASYNC_STORE
typedef int v2i_t __attribute__((ext_vector_type(2)));
typedef __attribute__((address_space(1))) v2i_t* gptr_v2i;  // global int2*
typedef __attribute__((address_space(3))) v2i_t* lptr_v2i;  // LDS int2*
#endif

static __device__ __forceinline__ void wait_dscnt0() {
#if defined(__has_builtin) && __has_builtin(__builtin_amdgcn_s_wait_dscnt)
    __builtin_amdgcn_s_wait_dscnt(0);
#else
    asm volatile("s_wait_dscnt 0" ::: "memory");
#endif
}

static __device__ __forceinline__ void wait_asynccnt0() {
#if defined(__has_builtin) && __has_builtin(__builtin_amdgcn_s_wait_asynccnt)
    __builtin_amdgcn_s_wait_asynccnt(0);
#else
    asm volatile("s_wait_asynccnt 0" ::: "memory");
#endif
}

// One wave32 per bag: lane d holds dims [2d, 2d+1] of the 64-dim accumulator.
// Lanes 0..19 additionally perform the 4-way cache probe for the bag's 20 lookups.
__global__ __launch_bounds__(256)
void emb_cache_gather_kernel(const int*   __restrict__ idx0,
                             const int*   __restrict__ occ0,
                             const float* __restrict__ w0,
                             const float* __restrict__ big0,
                             const int*   __restrict__ idx1,
                             const int*   __restrict__ occ1,
                             const float* __restrict__ w1,
                             const float* __restrict__ big1,
                             float*       __restrict__ out)
{
    const int wid  = (int)((blockIdx.x * blockDim.x + threadIdx.x) >> 5); // 0..8191
    const int lane = (int)(threadIdx.x & 31);
    const int t    = wid >> 12;           // table 0/1
    const int b    = wid & (BATCH - 1);   // bag within table

    const int*   idxp = t ? idx1 : idx0;
    const int*   occp = t ? occ1 : occ0;
    const float* wp   = t ? w1   : w0;
    const float* bigp = t ? big1 : big0;

    // --- probe: lanes 0..19 resolve one lookup each ---
    // enc >= 0  : row index into w (way*CACHE_SIZE + set)
    // enc <  0  : ~enc is the index into big (cache miss -> aux row == big[idx])
    int enc = 0;
    if (lane < POOL) {
        const int idx = idxp[b * POOL + lane];
        const unsigned set = (unsigned)idx % (unsigned)CACHE_SIZE;
        const int4 ways = *(const int4*)(occp + (size_t)set * NUM_WAYS); // b128: all 4 ways
        if      (ways.x == idx) enc = (int)(set);
        else if (ways.y == idx) enc = (int)(set + 1u * (unsigned)CACHE_SIZE);
        else if (ways.z == idx) enc = (int)(set + 2u * (unsigned)CACHE_SIZE);
        else if (ways.w == idx) enc = (int)(set + 3u * (unsigned)CACHE_SIZE);
        else                    enc = ~idx;
    }

    // --- gather + segment sum: 20 coalesced 256B row reads, fully unrolled ---
    float2 acc; acc.x = 0.0f; acc.y = 0.0f;
#pragma unroll
    for (int p = 0; p < POOL; ++p) {
        const int e = __builtin_amdgcn_readlane(enc, p);   // wave-uniform -> SALU addressing
        const float* row = (e >= 0) ? (wp   + (size_t)(unsigned)e    * DIM)
                                    : (bigp + (size_t)(unsigned)(~e) * DIM);
        const float2 v = ((const float2*)row)[lane];        // saddr-form global_load_b64
        acc.x += v.x;
        acc.y += v.y;
    }

    float2* orow = (float2*)(out + (((size_t)t * BATCH) + (size_t)b) * DIM);

#if HAVE_ASYNC_STORE
    // Stage in LDS and let the async engine drain the result (gfx1250 ASYNCcnt path).
    __shared__ float2 stage[256];
    stage[threadIdx.x] = acc;
    wait_dscnt0();                       // LDS write visible before async engine reads it
    __builtin_amdgcn_global_store_async_from_lds_b64(
        (gptr_v2i)((v2i_t*)(orow + lane)),
        (lptr_v2i)((v2i_t*)&stage[threadIdx.x]),
        0, 0);
    wait_asynccnt0();
#else
    orow[lane] = acc;                                       // global_store_b64
#endif
}

// d_in order (setup_inputs dict order):
//  [0] lS_i_0  [1] lS_o_0  [2] occ_0  [3] w_0  [4] big_0
//  [5] lS_i_1  [6] lS_o_1  [7] occ_1  [8] w_1  [9] big_1
// Integer inputs treated as int32 per harness convention.
// Output: (2, 4096, 64) float32, fully written by the kernel.
extern "C" void kernel_launch(void* const* d_in, const int* in_sizes, int n_in,
                              void* d_out, int out_size, void* d_ws, size_t ws_size,
                              hipStream_t stream) {
    (void)in_sizes; (void)n_in; (void)out_size; (void)d_ws; (void)ws_size;

    const int*   idx0 = (const int*)  d_in[0];
    const int*   occ0 = (const int*)  d_in[2];
    const float* w0   = (const float*)d_in[3];
    const float* big0 = (const float*)d_in[4];
    const int*   idx1 = (const int*)  d_in[5];
    const int*   occ1 = (const int*)  d_in[7];
    const float* w1   = (const float*)d_in[8];
    const float* big1 = (const float*)d_in[9];
    float* out = (float*)d_out;

    // 2 tables * 4096 bags = 8192 waves; 8 waves/block -> 1024 blocks of 256.
    const int total_waves = NTAB * BATCH;
    dim3 grid((unsigned)(total_waves / 8));
    dim3 block(256);
    emb_cache_gather_kernel<<<grid, block, 0, stream>>>(
        idx0, occ0, w0, big0, idx1, occ1, w1, big1, out);
}